// NLF_88708254531700
// MI455X (gfx1250) — compile-verified
//
#include <hip/hip_runtime.h>

typedef float v2f __attribute__((ext_vector_type(2)));
typedef float v8f __attribute__((ext_vector_type(8)));

static constexpr int Hh = 48, Ww = 96, HW = Hh * Ww;   // spatial dims
static constexpr int CHANNELS = 48 * 96;               // 4608 flattened channels

// ---------------------------------------------------------------------------
// Stage 0: per-direction L1-normalized guide coefficients in canonical layout.
// kcan[dir][j][r*Cc + c], j=0..3 are the x / prev-row taps, j=4 is the in-row
// recurrence coefficient 'a' (zeroed at canonical c==0, as in the reference).
// ---------------------------------------------------------------------------
__global__ __launch_bounds__(256) void prep_kcan(const float* __restrict__ g,
                                                 float* __restrict__ kcan) {
  int tid = blockIdx.x * blockDim.x + threadIdx.x;
  if (tid >= 4 * HW) return;
  int dir = tid / HW, p = tid - dir * HW;
  int Cc = (dir < 2) ? Ww : Hh;
  int r = p / Cc, c = p - r * Cc;
  int idx;
  if (dir == 0)      idx = r * Ww + c;                 // down
  else if (dir == 1) idx = (HW - 1) - r * Ww - c;      // up   (flip h,w)
  else if (dir == 2) idx = c * Ww + r;                 // right (transpose)
  else               idx = (HW - 1) - c * Ww - r;      // left (transpose+flip)
  float v[5], s = 0.f;
#pragma unroll
  for (int j = 0; j < 5; ++j) { v[j] = g[(dir * 5 + j) * HW + idx]; s += fabsf(v[j]); }
  float inv = 1.f / fmaxf(s, 1e-12f);
#pragma unroll
  for (int j = 0; j < 4; ++j) kcan[(dir * 5 + j) * HW + p] = v[j] * inv;
  float a = v[4] * inv;
  if (c == 0) a = 0.f;                                  // no left neighbor at row start
  kcan[(dir * 5 + 4) * HW + p] = a;
}

// ---------------------------------------------------------------------------
// Stage 1: build per-tile upper-triangular scan matrices L(j,c) = prod a(j+1..c)
// stored directly in WMMA B-operand lane layout (f32 4x16 per K-chunk):
//   chunk kk, lane l: {L(4kk + 2*(l>=16) + 0, l&15), L(4kk + 2*(l>=16) + 1, l&15)}
// One wave per 16x16 tile; 288 tiles per direction (48*6 == 96*3).
// ---------------------------------------------------------------------------
__global__ __launch_bounds__(128) void build_L(const float* __restrict__ kcan,
                                               float* __restrict__ Lbuf) {
  int gw = (blockIdx.x * blockDim.x + threadIdx.x) >> 5;
  int lane = threadIdx.x & 31;
  if (gw >= 4 * 288) return;
  int dir = gw / 288, ti = gw - dir * 288;
  int T  = (dir < 2) ? 6 : 3;
  int Cc = (dir < 2) ? Ww : Hh;
  int r = ti / T, t = ti - r * T;
  const float* kA = kcan + (dir * 5 + 4) * HW;
  float a[16];
#pragma unroll
  for (int i = 0; i < 16; ++i) a[i] = kA[r * Cc + t * 16 + i];
  int c = lane & 15, half = lane >> 4;
  v2f* out = reinterpret_cast<v2f*>(Lbuf) + (size_t)gw * 128;
#pragma unroll
  for (int kk = 0; kk < 4; ++kk) {
    int j0 = kk * 4 + half * 2;
    v2f v;
    {
      float p1 = 0.f;
      if (j0 <= c) { p1 = 1.f; for (int i = j0 + 1; i <= c; ++i) p1 *= a[i]; }
      v.x = p1;
    }
    {
      int j1 = j0 + 1; float p1 = 0.f;
      if (j1 <= c) { p1 = 1.f; for (int i = j1 + 1; i <= c; ++i) p1 *= a[i]; }
      v.y = p1;
    }
    out[kk * 32 + lane] = v;
  }
}

// ---------------------------------------------------------------------------
// Directional pass. ONE wave per workgroup owns 16 channels and walks canonical
// rows r=0..R-1 (single-wave WG => hardware barriers are S_NOP; __syncthreads
// only contributes the required s_wait_dscnt for intra-wave LDS ordering).
// x rows are staged memory->LDS with GLOBAL_LOAD_ASYNC_TO_LDS_B32 (ASYNCcnt).
// Per 16-wide tile: b(ch,c) = k0*x + k1*prev(c-1) + k2*prev(c) + k3*prev(c+1),
// carry folded into the tile's first column, then Out = B_tile x L via four
// chained V_WMMA_F32_16X16X4_F32 (K accumulation in f32).
// ---------------------------------------------------------------------------
__global__ __launch_bounds__(32) void nlf_pass(
    const float* __restrict__ src, float* __restrict__ dst,
    const float* __restrict__ kcan_dir, const float* __restrict__ Lbuf_dir,
    int R, int Cc, int T, int base, int sr, int sc) {
  __shared__ float sX[16 * 96];        // staged x row
  __shared__ float sP[2][16 * 98];     // prev/curr out rows, zero padded
  __shared__ float sK[4][96];          // coefficient row
  const int lane = threadIdx.x;        // 32 threads == 1 wave
  const int cb   = blockIdx.x * 16;    // channel base for this wave
  const int ps   = Cc + 2;             // padded row stride
  const float* kA = kcan_dir + 4 * HW;

  for (int i = lane; i < 2 * 16 * 98; i += 32) (&sP[0][0])[i] = 0.f;
  __syncthreads();

  const int m    = lane & 15;          // channel row within tile / column id
  const int half = lane >> 4;
  int pb = 0;
  for (int r = 0; r < R; ++r) {
    // stage this wave's 16-channel x row: async memory->LDS (no VGPR bounce)
    for (int ch = 0; ch < 16; ++ch)
      for (int c = lane; c < Cc; c += 32) {
        const float* gp = src + (size_t)(cb + ch) * HW + (base + r * sr + c * sc);
        unsigned la = (unsigned)(uintptr_t)&sX[ch * Cc + c];
        asm volatile("global_load_async_to_lds_b32 %0, %1, off"
                     :: "v"(la), "v"(gp) : "memory");
      }
    // stage coefficient row while the async DMA is in flight
    for (int j = 0; j < 4; ++j)
      for (int c = lane; c < Cc; c += 32)
        sK[j][c] = kcan_dir[j * HW + r * Cc + c];
    if (r + 1 < R) {   // warm caches for the next row (global_prefetch_b8)
      const float* pf = src + (size_t)(cb + m) * HW +
                        (base + (r + 1) * sr + (half * (Cc >> 1)) * sc);
      __builtin_prefetch(pf, 0, 0);
    }
    asm volatile("s_wait_asynccnt 0x0" ::: "memory");
    __syncthreads();                   // drain dscnt; barrier is S_NOP (1 wave)

    float* prev = sP[pb];
    float* curr = sP[pb ^ 1];
    for (int t = 0; t < T; ++t) {
      const int t0 = t * 16;
      const float a0 = kA[r * Cc + t0];
      const v2f* Lt = reinterpret_cast<const v2f*>(Lbuf_dir) + (size_t)(r * T + t) * 128;
      v8f acc = {0.f, 0.f, 0.f, 0.f, 0.f, 0.f, 0.f, 0.f};
#pragma unroll
      for (int kk = 0; kk < 4; ++kk) {
        const int j0 = kk * 4 + half * 2;   // A-operand K layout: 0/1 lo-half, 2/3 hi-half
        const int c0 = t0 + j0;
        float b0 = sK[0][c0]     * sX[m * Cc + c0]
                 + sK[1][c0]     * prev[m * ps + c0]
                 + sK[2][c0]     * prev[m * ps + c0 + 1]
                 + sK[3][c0]     * prev[m * ps + c0 + 2];
        float b1 = sK[0][c0 + 1] * sX[m * Cc + c0 + 1]
                 + sK[1][c0 + 1] * prev[m * ps + c0 + 1]
                 + sK[2][c0 + 1] * prev[m * ps + c0 + 2]
                 + sK[3][c0 + 1] * prev[m * ps + c0 + 3];
        if (kk == 0) {                      // fold inter-tile carry into first column
          float cf = (half == 0) ? a0 : 0.f;
          b0 += cf * curr[m * ps + t0];     // slot t0 == column t0-1 (zero pad at c=-1)
        }
        v2f A; A.x = b0; A.y = b1;
        v2f B = Lt[kk * 32 + lane];
        acc = __builtin_amdgcn_wmma_f32_16x16x4_f32(false, A, false, B, (short)0, acc,
                                                    false, false);
      }
      // C/D layout: lane -> column t0 + (lane&15); vgpr i -> channel 8*half + i
      const int c = t0 + m;
      const int chb = half * 8;
      const int gidx = base + r * sr + c * sc;
#pragma unroll
      for (int i = 0; i < 8; ++i) {
        float v = acc[i];
        curr[(chb + i) * ps + (c + 1)] = v;
        dst[(size_t)(cb + chb + i) * HW + gidx] = v;
      }
      __syncthreads();                      // publish carry column (dscnt drain only)
    }
    pb ^= 1;                                // curr becomes prev
  }
}

// ---------------------------------------------------------------------------
extern "C" void kernel_launch(void* const* d_in, const int* /*in_sizes*/, int /*n_in*/,
                              void* d_out, int /*out_size*/, void* d_ws, size_t /*ws_size*/,
                              hipStream_t stream) {
  const float* x = (const float*)d_in[0];     // (1,48,96,48,96) -> (4608, 48, 96)
  const float* g = (const float*)d_in[1];     // (1,20,48,96)
  float* out  = (float*)d_out;
  float* kcan = (float*)d_ws;                 // 4*5*HW floats   (~368 KB)
  float* Lbuf = kcan + 4 * 5 * HW;            // 4*288*256 floats (~1.18 MB)

  prep_kcan<<<(4 * HW + 255) / 256, 256, 0, stream>>>(g, kcan);
  build_L<<<(4 * 288 * 32) / 128, 128, 0, stream>>>(kcan, Lbuf);

  // canonical geometry per direction: idx = base + r*sr + c*sc
  const int Rr[4] = {48, 48, 96, 96};
  const int Cc[4] = {96, 96, 48, 48};
  const int Tt[4] = {6, 6, 3, 3};
  const int Bb[4] = {0, HW - 1, 0, HW - 1};
  const int Sr[4] = {96, -96, 1, -1};
  const int Sc[4] = {1, -1, 96, -96};
  const float* srcs[4] = {x, out, out, out};  // pass 1 reads x; 2-4 run in place on out

  const int nblocks = CHANNELS / 16;          // 288 single-wave workgroups
  for (int d = 0; d < 4; ++d)
    nlf_pass<<<nblocks, 32, 0, stream>>>(
        srcs[d], out, kcan + d * 5 * HW, Lbuf + (size_t)d * 288 * 256,
        Rr[d], Cc[d], Tt[d], Bb[d], Sr[d], Sc[d]);
}